// QuantumFourierAttention_86973087744618
// MI455X (gfx1250) — compile-verified
//
#include <hip/hip_runtime.h>
#include <hip/hip_bf16.h>
#include <math.h>

// Problem constants
#define BB 2
#define NN 2048
#define DD 512
#define HH 8
#define DH 64
#define BH (BB*HH)              // 16
#define SCALE 0.125f            // 1/sqrt(64)
#define KC 64                   // k-chunk for staged GEMMs (== DH)
#define WPAD 68                 // padded LDS row stride (floats) -> conflict-free b64 reads

typedef float v2f __attribute__((ext_vector_type(2)));
typedef float v8f __attribute__((ext_vector_type(8)));
typedef int   v4i __attribute__((ext_vector_type(4)));

// fp32 WMMA 16x16x4: D = A(16x4) * B(4x16) + C(16x16)
static __device__ inline v8f wmma4(v2f a, v2f b, v8f c) {
    return __builtin_amdgcn_wmma_f32_16x16x4_f32(
        /*neg_a=*/false, a, /*neg_b=*/false, b,
        /*c_mod=*/(short)0, c, /*reuse_a=*/false, /*reuse_b=*/false);
}

// ---------------------------------------------------------------------------
// CDNA5 async global->LDS staging (ASYNCcnt path), with portable fallback.
// ---------------------------------------------------------------------------
#if defined(__HIP_DEVICE_COMPILE__) && \
    __has_builtin(__builtin_amdgcn_global_load_async_to_lds_b128) && \
    __has_builtin(__builtin_amdgcn_s_wait_asynccnt)
#define HAVE_ASYNC 1
#else
#define HAVE_ASYNC 0
#endif

#if HAVE_ASYNC
typedef __attribute__((address_space(1))) v4i g_v4i;   // global 16B vector
typedef __attribute__((address_space(3))) v4i l_v4i;   // LDS 16B vector
#endif

// copy 16 bytes (per lane) global -> LDS
static __device__ inline void stage16(const float* g, float* l) {
#if HAVE_ASYNC
    // inttoptr casts: AS1 address == 64-bit VA; AS3 address == low 32 bits of
    // the generic LDS pointer.
    __builtin_amdgcn_global_load_async_to_lds_b128(
        (g_v4i*)(unsigned long long)g,
        (l_v4i*)(unsigned)(unsigned long long)l,
        0, 0);
#else
    float4 v = *(const float4*)g;
    *(float4*)l = v;
#endif
}
static __device__ inline void stage_drain_keep2() {
#if HAVE_ASYNC
    __builtin_amdgcn_s_wait_asynccnt(2);
#endif
}
static __device__ inline void stage_drain_all() {
#if HAVE_ASYNC
    __builtin_amdgcn_s_wait_asynccnt(0);
#endif
}

// ---------------------------------------------------------------------------
// Kernel 1: complex projection X(B*N,D) @ W^T + b -> head layout (BH,N,DH)
// real = Xr Wr^T - Xi Wi^T + br ;  imag = Xr Wi^T + Xi Wr^T + bi
// Block = 128 rows x 16 cols (8 waves, one 16x16 complex tile each).
// Weight k-panel (16n x 64k, r+i) double-buffered in LDS via async copies.
// ---------------------------------------------------------------------------
__global__ void cproj_qkv(const float* __restrict__ Xr, const float* __restrict__ Xi,
                          const float* __restrict__ Wr, const float* __restrict__ Wi,
                          const float* __restrict__ br, const float* __restrict__ bi,
                          float* __restrict__ Yr, float* __restrict__ Yi)
{
    __shared__ float sWr[2][16][WPAD];
    __shared__ float sWi[2][16][WPAD];

    const int tid  = threadIdx.x;
    const int lane = tid & 31;
    const int wave = tid >> 5;
    const int mb = blockIdx.x >> 5;              // 32 row-blocks of 128
    const int nb = blockIdx.x & 31;              // 32 col-blocks of 16
    const int row0 = mb * 128 + wave * 16;
    const int col0 = nb * 16;
    const int m = row0 + (lane & 15);
    const int n = col0 + (lane & 15);
    const int kb = (lane >> 4) << 1;             // 0 or 2

    // staging coordinates: thread t copies row tn, 4 floats at tk4
    const int tn  = tid >> 4;                    // 0..15
    const int tk4 = (tid & 15) << 2;             // 0..60

    const float* xr = Xr + (size_t)m * DD + kb;
    const float* xi = Xi + (size_t)m * DD + kb;
    const float* wrow_r = Wr + (size_t)(col0 + tn) * DD + tk4;
    const float* wrow_i = Wi + (size_t)(col0 + tn) * DD + tk4;

    // prologue: stage chunk 0
    stage16(wrow_r, &sWr[0][tn][tk4]);
    stage16(wrow_i, &sWi[0][tn][tk4]);

    v8f accR = {}; v8f accI = {};
    for (int c = 0; c < DD / KC; ++c) {
        if (c < DD / KC - 1) {
            stage16(wrow_r + (c + 1) * KC, &sWr[(c + 1) & 1][tn][tk4]);
            stage16(wrow_i + (c + 1) * KC, &sWi[(c + 1) & 1][tn][tk4]);
            stage_drain_keep2();
        } else {
            stage_drain_all();
        }
        __syncthreads();

        const float* bwrp = &sWr[c & 1][lane & 15][kb];
        const float* bwip = &sWi[c & 1][lane & 15][kb];
        const int kbase = c * KC;
        for (int k = 0; k < KC; k += 4) {
            v2f ar;  ar.x  = xr[kbase + k]; ar.y  = xr[kbase + k + 1];
            v2f ai;  ai.x  = xi[kbase + k]; ai.y  = xi[kbase + k + 1];
            v2f nai; nai.x = -ai.x;         nai.y = -ai.y;
            v2f bwr; bwr.x = bwrp[k];       bwr.y = bwrp[k + 1];
            v2f bwi; bwi.x = bwip[k];       bwi.y = bwip[k + 1];
            accR = wmma4(ar,  bwr, accR);
            accR = wmma4(nai, bwi, accR);
            accI = wmma4(ar,  bwi, accI);
            accI = wmma4(ai,  bwr, accI);
        }
        __syncthreads();
    }

    const float biasr = br[n], biasi = bi[n];
    const int h = n >> 6, dh = n & 63;
    const int rhalf = (lane >> 4) << 3;
    for (int v = 0; v < 8; ++v) {
        int row = row0 + v + rhalf;
        int b = row >> 11, nn = row & (NN - 1);
        size_t idx = (((size_t)(b * HH + h) * NN) + nn) * DH + dh;
        Yr[idx] = accR[v] + biasr;
        Yi[idx] = accI[v] + biasi;
    }
}

// ---------------------------------------------------------------------------
// Kernel 2: 2048-point radix-2 complex FFT along n, one (bh,dh) line per block.
// In-place in LDS with bit-reversal. inverse!=0 => conj twiddles + 1/N scale.
// ---------------------------------------------------------------------------
__global__ void fft_kernel(float* __restrict__ Re, float* __restrict__ Im, int inverse)
{
    __shared__ float sr[NN];
    __shared__ float si[NN];
    const int line = blockIdx.x;          // 0..1023
    const int bh = line >> 6, dh = line & 63;
    const int tid = threadIdx.x;          // 256 threads
    const size_t base = ((size_t)bh * NN) * DH + dh;

    for (int t = 0; t < 8; ++t) {
        int i = t * 256 + tid;
        int j = (int)(__brev((unsigned)i) >> 21);   // 11-bit reversal
        sr[i] = Re[base + (size_t)j * DH];
        si[i] = Im[base + (size_t)j * DH];
    }
    __syncthreads();

    const float sign = inverse ? 1.0f : -1.0f;
    for (int len = 2; len <= NN; len <<= 1) {
        int half = len >> 1;
        for (int t = 0; t < 4; ++t) {
            int bidx = t * 256 + tid;               // 1024 butterflies / stage
            int grp = bidx / half;
            int pos = bidx - grp * half;
            int i0 = grp * len + pos;
            int i1 = i0 + half;
            float ang = sign * 6.2831853071795865f * (float)pos / (float)len;
            float s, c;
            __sincosf(ang, &s, &c);
            float xr = sr[i1], xi = si[i1];
            float tr = xr * c - xi * s;
            float ti = xr * s + xi * c;
            float ur = sr[i0], ui = si[i0];
            sr[i0] = ur + tr; si[i0] = ui + ti;
            sr[i1] = ur - tr; si[i1] = ui - ti;
        }
        __syncthreads();
    }

    const float scl = inverse ? (1.0f / (float)NN) : 1.0f;
    for (int t = 0; t < 8; ++t) {
        int i = t * 256 + tid;
        Re[base + (size_t)i * DH] = sr[i] * scl;
        Im[base + (size_t)i * DH] = si[i] * scl;
    }
}

// ---------------------------------------------------------------------------
// Kernel 3 (pass A): per-row softmax stats over scores.real = Qr.Kr + Qi.Ki.
// ---------------------------------------------------------------------------
__global__ void attn_stats(const float* __restrict__ Qr, const float* __restrict__ Qi,
                           const float* __restrict__ Kr, const float* __restrict__ Ki,
                           float* __restrict__ Mrow, float* __restrict__ Lrow)
{
    const int lane = threadIdx.x & 31;
    const int wave = threadIdx.x >> 5;
    const int bh = blockIdx.x >> 7;
    const int i0 = (blockIdx.x & 127) << 4;
    const size_t base = (size_t)bh * NN * DH;
    const int kb = (lane >> 4) << 1;
    const int mr = i0 + (lane & 15);

    const float* qrp = Qr + base + (size_t)mr * DH + kb;
    const float* qip = Qi + base + (size_t)mr * DH + kb;

    float mloc[8], lloc[8];
    for (int v = 0; v < 8; ++v) { mloc[v] = -3.0e38f; lloc[v] = 0.0f; }

    for (int jt = wave; jt < 128; jt += 8) {
        int ncol = (jt << 4) + (lane & 15);
        const float* krp = Kr + base + (size_t)ncol * DH + kb;   // B[k][j] = K[j][k]
        const float* kip = Ki + base + (size_t)ncol * DH + kb;
        v8f acc = {};
        for (int k = 0; k < DH; k += 4) {
            v2f a1; a1.x = qrp[k]; a1.y = qrp[k + 1];
            v2f b1; b1.x = krp[k]; b1.y = krp[k + 1];
            acc = wmma4(a1, b1, acc);
            v2f a2; a2.x = qip[k]; a2.y = qip[k + 1];
            v2f b2; b2.x = kip[k]; b2.y = kip[k + 1];
            acc = wmma4(a2, b2, acc);
        }
        for (int v = 0; v < 8; ++v) {
            float s = acc[v] * SCALE;
            float tmax = s;
            for (int off = 1; off < 16; off <<= 1)
                tmax = fmaxf(tmax, __shfl_xor(tmax, off, 32));
            float mnew = fmaxf(mloc[v], tmax);
            float p = __expf(s - mnew);
            float psum = p;
            for (int off = 1; off < 16; off <<= 1)
                psum += __shfl_xor(psum, off, 32);
            lloc[v] = lloc[v] * __expf(mloc[v] - mnew) + psum;
            mloc[v] = mnew;
        }
    }

    __shared__ float sm[8][16];
    __shared__ float sl[8][16];
    if (lane == 0 || lane == 16) {
        int rb = (lane >> 4) << 3;
        for (int v = 0; v < 8; ++v) { sm[wave][rb + v] = mloc[v]; sl[wave][rb + v] = lloc[v]; }
    }
    __syncthreads();
    if (threadIdx.x < 16) {
        int r = threadIdx.x;
        float M = -3.0e38f;
        for (int w = 0; w < 8; ++w) M = fmaxf(M, sm[w][r]);
        float L = 0.0f;
        for (int w = 0; w < 8; ++w) L += sl[w][r] * __expf(sm[w][r] - M);
        Mrow[(size_t)bh * NN + i0 + r] = M;
        Lrow[(size_t)bh * NN + i0 + r] = fmaxf(L, 1e-30f);
    }
}

// ---------------------------------------------------------------------------
// Kernel 4 (pass B): out_f = softmax(scores) @ Vf (real attn x complex V).
// ---------------------------------------------------------------------------
__global__ void attn_out(const float* __restrict__ Qr, const float* __restrict__ Qi,
                         const float* __restrict__ Kr, const float* __restrict__ Ki,
                         const float* __restrict__ Vr, const float* __restrict__ Vi,
                         const float* __restrict__ Mrow, const float* __restrict__ Lrow,
                         float* __restrict__ Or, float* __restrict__ Oi)
{
    __shared__ float sp[8][16][17];        // per-wave P staging (padded)
    __shared__ float soacc[8][16][16];     // 8 output tiles (Vr n0..3, Vi n0..3)

    const int lane = threadIdx.x & 31;
    const int wave = threadIdx.x >> 5;
    const int bh = blockIdx.x >> 7;
    const int i0 = (blockIdx.x & 127) << 4;
    const size_t base = (size_t)bh * NN * DH;
    const int kb = (lane >> 4) << 1;
    const int mr = i0 + (lane & 15);
    const int rhalf = (lane >> 4) << 3;

    for (int t = threadIdx.x; t < 8 * 256; t += 256) ((float*)soacc)[t] = 0.0f;
    __syncthreads();

    float mstat[8], linv[8];
    for (int v = 0; v < 8; ++v) {
        int r = i0 + v + rhalf;
        mstat[v] = Mrow[(size_t)bh * NN + r];
        linv[v]  = 1.0f / Lrow[(size_t)bh * NN + r];
    }

    const float* qrp = Qr + base + (size_t)mr * DH + kb;
    const float* qip = Qi + base + (size_t)mr * DH + kb;

    v8f zero = {};
    v8f accO[8];
    for (int t = 0; t < 8; ++t) accO[t] = zero;

    for (int jt = wave; jt < 128; jt += 8) {
        int j0 = jt << 4;
        int ncol = j0 + (lane & 15);
        const float* krp = Kr + base + (size_t)ncol * DH + kb;
        const float* kip = Ki + base + (size_t)ncol * DH + kb;
        v8f s = {};
        for (int k = 0; k < DH; k += 4) {
            v2f a1; a1.x = qrp[k]; a1.y = qrp[k + 1];
            v2f b1; b1.x = krp[k]; b1.y = krp[k + 1];
            s = wmma4(a1, b1, s);
            v2f a2; a2.x = qip[k]; a2.y = qip[k + 1];
            v2f b2; b2.x = kip[k]; b2.y = kip[k + 1];
            s = wmma4(a2, b2, s);
        }
        for (int v = 0; v < 8; ++v) {
            float p = __expf(s[v] * SCALE - mstat[v]) * linv[v];
            sp[wave][v + rhalf][lane & 15] = p;
        }
        __syncthreads();
        // P(16x16) @ V(16x64) for real and imag V
        for (int kk = 0; kk < 16; kk += 4) {
            v2f ap;
            ap.x = sp[wave][lane & 15][kk + kb];
            ap.y = sp[wave][lane & 15][kk + kb + 1];
            const float* vr0 = Vr + base + (size_t)(j0 + kk + kb) * DH;
            const float* vi0 = Vi + base + (size_t)(j0 + kk + kb) * DH;
            for (int nt = 0; nt < 4; ++nt) {
                int n = (nt << 4) + (lane & 15);
                v2f bv; bv.x = vr0[n]; bv.y = vr0[DH + n];
                accO[nt] = wmma4(ap, bv, accO[nt]);
                v2f bw; bw.x = vi0[n]; bw.y = vi0[DH + n];
                accO[4 + nt] = wmma4(ap, bw, accO[4 + nt]);
            }
        }
        __syncthreads();
    }

    for (int t = 0; t < 8; ++t)
        for (int v = 0; v < 8; ++v)
            atomicAdd(&soacc[t][v + rhalf][lane & 15], accO[t][v]);
    __syncthreads();

    for (int t = threadIdx.x; t < 2048; t += 256) {
        int nt  = t >> 8;
        int row = (t >> 4) & 15;
        int col = t & 15;
        float val = ((float*)soacc)[t];
        size_t idx = base + (size_t)(i0 + row) * DH + ((nt & 3) << 4) + col;
        if (nt < 4) Or[idx] = val; else Oi[idx] = val;
    }
}

// ---------------------------------------------------------------------------
// Kernel 5: output projection from head-layout (BH,N,DH) complex input to
// interleaved complex64 (B,N,D) in d_out. Same LDS-staged weight panel; the
// 64-wide k-chunk aligns with head boundaries so the A gather is contiguous.
// ---------------------------------------------------------------------------
__global__ void cproj_out(const float* __restrict__ Ar, const float* __restrict__ Ai,
                          const float* __restrict__ Wr, const float* __restrict__ Wi,
                          const float* __restrict__ br, const float* __restrict__ bi,
                          float* __restrict__ out)
{
    __shared__ float sWr[2][16][WPAD];
    __shared__ float sWi[2][16][WPAD];

    const int tid  = threadIdx.x;
    const int lane = tid & 31;
    const int wave = tid >> 5;
    const int mb = blockIdx.x >> 5;
    const int nb = blockIdx.x & 31;
    const int row0 = mb * 128 + wave * 16;
    const int col0 = nb * 16;
    const int m = row0 + (lane & 15);
    const int n = col0 + (lane & 15);
    const int kb = (lane >> 4) << 1;

    const int tn  = tid >> 4;
    const int tk4 = (tid & 15) << 2;

    const int b = m >> 11, nn = m & (NN - 1);
    const size_t abase = ((size_t)b * HH) * NN * DH + (size_t)nn * DH;
    const float* wrow_r = Wr + (size_t)(col0 + tn) * DD + tk4;
    const float* wrow_i = Wi + (size_t)(col0 + tn) * DD + tk4;

    stage16(wrow_r, &sWr[0][tn][tk4]);
    stage16(wrow_i, &sWi[0][tn][tk4]);

    v8f accR = {}; v8f accI = {};
    for (int c = 0; c < DD / KC; ++c) {         // chunk c == head h
        if (c < DD / KC - 1) {
            stage16(wrow_r + (c + 1) * KC, &sWr[(c + 1) & 1][tn][tk4]);
            stage16(wrow_i + (c + 1) * KC, &sWi[(c + 1) & 1][tn][tk4]);
            stage_drain_keep2();
        } else {
            stage_drain_all();
        }
        __syncthreads();

        const float* arp = Ar + abase + (size_t)c * (NN * DH) + kb;  // head c, dh = k+kb
        const float* aip = Ai + abase + (size_t)c * (NN * DH) + kb;
        const float* bwrp = &sWr[c & 1][lane & 15][kb];
        const float* bwip = &sWi[c & 1][lane & 15][kb];
        for (int k = 0; k < KC; k += 4) {
            v2f ar;  ar.x  = arp[k];  ar.y  = arp[k + 1];
            v2f ai;  ai.x  = aip[k];  ai.y  = aip[k + 1];
            v2f nai; nai.x = -ai.x;   nai.y = -ai.y;
            v2f bwr; bwr.x = bwrp[k]; bwr.y = bwrp[k + 1];
            v2f bwi; bwi.x = bwip[k]; bwi.y = bwip[k + 1];
            accR = wmma4(ar,  bwr, accR);
            accR = wmma4(nai, bwi, accR);
            accI = wmma4(ar,  bwi, accI);
            accI = wmma4(ai,  bwr, accI);
        }
        __syncthreads();
    }

    const float biasr = br[n], biasi = bi[n];
    const int rhalf = (lane >> 4) << 3;
    for (int v = 0; v < 8; ++v) {
        int row = row0 + v + rhalf;
        size_t o = ((size_t)row * DD + n) * 2;
        out[o]     = accR[v] + biasr;   // complex64: real
        out[o + 1] = accI[v] + biasi;   // complex64: imag
    }
}

// ---------------------------------------------------------------------------
extern "C" void kernel_launch(void* const* d_in, const int* in_sizes, int n_in,
                              void* d_out, int out_size, void* d_ws, size_t ws_size,
                              hipStream_t stream) {
    (void)in_sizes; (void)n_in; (void)out_size; (void)ws_size;

    const float* xr = (const float*)d_in[0];
    const float* xi = (const float*)d_in[1];
    const float* qWr = (const float*)d_in[2];  const float* qWi = (const float*)d_in[3];
    const float* qbr = (const float*)d_in[4];  const float* qbi = (const float*)d_in[5];
    const float* kWr = (const float*)d_in[6];  const float* kWi = (const float*)d_in[7];
    const float* kbr = (const float*)d_in[8];  const float* kbi = (const float*)d_in[9];
    const float* vWr = (const float*)d_in[10]; const float* vWi = (const float*)d_in[11];
    const float* vbr = (const float*)d_in[12]; const float* vbi = (const float*)d_in[13];
    const float* oWr = (const float*)d_in[14]; const float* oWi = (const float*)d_in[15];
    const float* obr = (const float*)d_in[16]; const float* obi = (const float*)d_in[17];

    const size_t T = (size_t)BH * NN * DH;   // 2,097,152 floats per tensor
    float* w  = (float*)d_ws;
    float* Qr = w;        float* Qi = Qr + T;
    float* Kr = Qi + T;   float* Ki = Kr + T;
    float* Vr = Ki + T;   float* Vi = Vr + T;
    float* Or = Vi + T;   float* Oi = Or + T;
    float* Mrow = Oi + T; float* Lrow = Mrow + (size_t)BH * NN;

    // 1) complex projections -> head layout (128x16 block tiles: 32x32 blocks)
    cproj_qkv<<<1024, 256, 0, stream>>>(xr, xi, qWr, qWi, qbr, qbi, Qr, Qi);
    cproj_qkv<<<1024, 256, 0, stream>>>(xr, xi, kWr, kWi, kbr, kbi, Kr, Ki);
    cproj_qkv<<<1024, 256, 0, stream>>>(xr, xi, vWr, vWi, vbr, vbi, Vr, Vi);

    // 2) forward FFT along sequence axis
    fft_kernel<<<1024, 256, 0, stream>>>(Qr, Qi, 0);
    fft_kernel<<<1024, 256, 0, stream>>>(Kr, Ki, 0);
    fft_kernel<<<1024, 256, 0, stream>>>(Vr, Vi, 0);

    // 3) softmax stats, 4) attention output (freq domain)
    attn_stats<<<2048, 256, 0, stream>>>(Qr, Qi, Kr, Ki, Mrow, Lrow);
    attn_out<<<2048, 256, 0, stream>>>(Qr, Qi, Kr, Ki, Vr, Vi, Mrow, Lrow, Or, Oi);

    // 5) inverse FFT
    fft_kernel<<<1024, 256, 0, stream>>>(Or, Oi, 1);

    // 6) output projection -> interleaved complex64
    cproj_out<<<1024, 256, 0, stream>>>(Or, Oi, oWr, oWi, obr, obi, (float*)d_out);
}